// MultiScalePeriodicAttentionLayer_19653770346958
// MI455X (gfx1250) — compile-verified
//
#include <hip/hip_runtime.h>
#include <cstddef>
#include <cstdint>

typedef __attribute__((ext_vector_type(2))) float v2f;
typedef __attribute__((ext_vector_type(8))) float v8f;

#define BB 16
#define TT 16
#define HH 32
#define WW 32
#define DD 64
#define NEXP 6
#define LN_EPS 1e-6f
#define COMBINE_EPS 2.2204460492503131e-16f
#define KB 32            // K block staged in LDS
#define APAD 34          // A LDS row stride (32 + 2, even => aligned b64 pairs)

// ---------------------------------------------------------------------------
// fp32 GEMM on the CDNA5 matrix pipe: C[M,N] = A[M,K] * B[K,N] (+bias[n]).
// Block = 128 threads (4 waves), 64x64 output tile. A(64xKB) staged row-major
// (even stride => fragment = one aligned ds_load_b64); B(KBx64) staged as
// K-pair-interleaved float2 (Bsp[p][n] = {B[2p][n], B[2p+1][n]}) so each B
// fragment is also a single conflict-free ds_load_b64. Per K-block we gather
// all fragments into registers first, then issue 32 back-to-back
// v_wmma_f32_16x16x4_f32 (one A fragment reused across 4 N-tiles).
// Requires M%64==0, N%64==0, K%32==0 (true for every GEMM in this layer).
// ---------------------------------------------------------------------------
__global__ __launch_bounds__(128)
void gemm_f32_wmma(const float* __restrict__ A, const float* __restrict__ B,
                   const float* __restrict__ bias, float* __restrict__ C,
                   int M, int N, int K) {
  __shared__ __align__(16) float As[64 * APAD];
  __shared__ __align__(16) v2f   Bsp[(KB / 2) * 64];   // [pair][n]

  const int tid  = threadIdx.x;
  const int lane = tid & 31;
  const int wave = tid >> 5;
  const int l15  = lane & 15;
  const int hi   = lane >> 4;           // 0 -> K {0,1}, 1 -> K {2,3}
  const int tn   = blockIdx.x * 64;
  const int tm   = blockIdx.y * 64;

  v8f acc[4] = {};

  const float* __restrict__ ap =
      &As[(wave * 16 + l15) * APAD + 2 * hi];
  float* __restrict__ bsf = (float*)Bsp;

  for (int k0 = 0; k0 < K; k0 += KB) {
    // ---- stage A (64 x 32) and B (32 x 64) into LDS, coalesced ----
#pragma unroll
    for (int r = 0; r < 16; ++r) {
      const int i = tid + 128 * r;      // 0..2047
      const int m = i >> 5, kk = i & 31;
      As[m * APAD + kk] = A[(size_t)(tm + m) * K + (k0 + kk)];
    }
#pragma unroll
    for (int r = 0; r < 16; ++r) {
      const int i = tid + 128 * r;      // 0..2047
      const int kk = i >> 6, n = i & 63;
      bsf[((kk >> 1) * 64 + n) * 2 + (kk & 1)] =
          B[(size_t)(k0 + kk) * N + (tn + n)];
    }
    __syncthreads();

    // ---- load phase: all fragments for this K-block into registers ----
    v2f afr[8];
    v2f bfr[8][4];
#pragma unroll
    for (int s = 0; s < 8; ++s) {       // k-step s covers K = 4s..4s+3
      afr[s] = *(const v2f*)(ap + 4 * s);                 // {k+2hi, k+2hi+1}
      const v2f* __restrict__ bpv = &Bsp[(2 * s + hi) * 64 + l15];
#pragma unroll
      for (int nt = 0; nt < 4; ++nt) {
        bfr[s][nt] = bpv[nt * 16];
      }
    }

    // ---- compute phase: 32 back-to-back WMMAs ----
#pragma unroll
    for (int s = 0; s < 8; ++s) {
#pragma unroll
      for (int nt = 0; nt < 4; ++nt) {
        acc[nt] = __builtin_amdgcn_wmma_f32_16x16x4_f32(
            false, afr[s], false, bfr[s][nt], (short)0, acc[nt], false, false);
      }
    }
    __syncthreads();
  }

  // ---- epilogue: C VGPR r -> M = r + 8*hi, N = l15 (per 16x16 tile) ----
  const int row0 = tm + wave * 16 + hi * 8;
#pragma unroll
  for (int nt = 0; nt < 4; ++nt) {
    const int col = tn + nt * 16 + l15;
    const float bv = bias ? bias[col] : 0.0f;
#pragma unroll
    for (int r = 0; r < 8; ++r) {
      C[(size_t)(row0 + r) * N + col] = acc[nt][r] + bv;
    }
  }
}

// ---------------------------------------------------------------------------
// im2col for 2x2 stride-2 VALID conv on NHWC activations.
// act: (BT, Si, Si, Ci) -> col: (BT*So*So, 4*Ci), k = (dy*2+dx)*Ci + ci
// ---------------------------------------------------------------------------
__global__ void im2col2x2(const float* __restrict__ act, float* __restrict__ col,
                          int BT, int Si, int Ci) {
  const int So = Si >> 1;
  const size_t Kc = 4 * (size_t)Ci;
  const size_t total = (size_t)BT * So * So * Kc;
  for (size_t idx = blockIdx.x * (size_t)blockDim.x + threadIdx.x; idx < total;
       idx += (size_t)gridDim.x * blockDim.x) {
    const size_t row = idx / Kc;
    const int k  = (int)(idx % Kc);
    const int ci = k % Ci;
    const int q  = k / Ci;
    const int dy = q >> 1, dx = q & 1;
    const int xo = (int)(row % So);
    const size_t r2 = row / So;
    const int yo = (int)(r2 % So);
    const int bt = (int)(r2 / So);
    col[idx] = act[(((size_t)bt * Si + (2 * yo + dy)) * Si + (2 * xo + dx)) * Ci + ci];
  }
}

// conv weight OIHW (Co,Ci,2,2) -> Bt (K=4Ci, Co) with k=(dy*2+dx)*Ci+ci
__global__ void conv_wtrans(const float* __restrict__ w, float* __restrict__ bt,
                            int Ci, int Co) {
  const size_t total = (size_t)Co * Ci * 4;
  for (size_t idx = blockIdx.x * (size_t)blockDim.x + threadIdx.x; idx < total;
       idx += (size_t)gridDim.x * blockDim.x) {
    const int n = (int)(idx % Co);
    const size_t k = idx / Co;
    const int ci = (int)(k % Ci);
    const int q  = (int)(k / Ci);
    bt[idx] = w[((size_t)n * Ci + ci) * 4 + q];
  }
}

// Per-row LayerNorm (over N channels) + exact GELU. in/out: (M, N) row-major.
__global__ __launch_bounds__(256)
void ln_gelu(const float* __restrict__ in, const float* __restrict__ g,
             const float* __restrict__ b, float* __restrict__ out, int N) {
  __shared__ float sh[512];
  const size_t row = blockIdx.x;
  const float* __restrict__ r = in + row * (size_t)N;
  float s = 0.0f, ss = 0.0f;
  for (int n = threadIdx.x; n < N; n += 256) {
    const float v = r[n];
    s += v; ss += v * v;
  }
  sh[threadIdx.x] = s;
  sh[256 + threadIdx.x] = ss;
  __syncthreads();
  for (int off = 128; off > 0; off >>= 1) {
    if ((int)threadIdx.x < off) {
      sh[threadIdx.x] += sh[threadIdx.x + off];
      sh[256 + threadIdx.x] += sh[256 + threadIdx.x + off];
    }
    __syncthreads();
  }
  const float mu  = sh[0] / N;
  const float var = sh[256] / N - mu * mu;
  const float inv = rsqrtf(var + LN_EPS);
  for (int n = threadIdx.x; n < N; n += 256) {
    const float y = (r[n] - mu) * inv * g[n] + b[n];
    out[row * (size_t)N + n] = 0.5f * y * (1.0f + erff(y * 0.70710678118654752f));
  }
}

// rFFT over T=16 (ortho), drop DC, |.| averaged over C=2048 channels.
// h: (B*T, 2048), amp: (B, 8). blockIdx.x = b, blockIdx.y = f-1.
__global__ __launch_bounds__(256)
void fft_amp(const float* __restrict__ h, float* __restrict__ amp) {
  __shared__ float sh[256];
  const int b = blockIdx.x;
  const int f = blockIdx.y + 1;
  const int C = 2048;
  float sum = 0.0f;
  for (int c = threadIdx.x; c < C; c += 256) {
    float re = 0.0f, im = 0.0f;
#pragma unroll
    for (int t = 0; t < TT; ++t) {
      const float ang = -2.0f * 3.14159265358979323846f * (float)(f * t) / 16.0f;
      const float v = h[(size_t)(b * TT + t) * C + c];
      re += v * cosf(ang);
      im += v * sinf(ang);
    }
    sum += sqrtf(re * re + im * im) * 0.25f;  // 1/sqrt(16) ortho norm
  }
  sh[threadIdx.x] = sum;
  __syncthreads();
  for (int off = 128; off > 0; off >>= 1) {
    if ((int)threadIdx.x < off) sh[threadIdx.x] += sh[threadIdx.x + off];
    __syncthreads();
  }
  if (threadIdx.x == 0) amp[b * 8 + blockIdx.y] = sh[0] / (float)C;
}

// logits = amp @ w_gate (8x6); top-2 softmax -> gates (B, 6). One thread per b.
__global__ void gate_topk(const float* __restrict__ amp, const float* __restrict__ wg,
                          float* __restrict__ gates) {
  const int b = threadIdx.x;
  if (b >= BB) return;
  float logit[NEXP];
  for (int e = 0; e < NEXP; ++e) {
    float s = 0.0f;
    for (int f = 0; f < 8; ++f) s += amp[b * 8 + f] * wg[f * NEXP + e];
    logit[e] = s;
  }
  int i1 = 0;
  for (int e = 1; e < NEXP; ++e) if (logit[e] > logit[i1]) i1 = e;
  int i2 = -1;
  for (int e = 0; e < NEXP; ++e) {
    if (e == i1) continue;
    if (i2 < 0 || logit[e] > logit[i2]) i2 = e;
  }
  const float m  = logit[i1];
  const float e1 = expf(logit[i1] - m);
  const float e2 = expf(logit[i2] - m);
  const float inv = 1.0f / (e1 + e2);
  for (int e = 0; e < NEXP; ++e) gates[b * NEXP + e] = 0.0f;
  gates[b * NEXP + i1] = e1 * inv;
  gates[b * NEXP + i2] = e2 * inv;
}

// Build segmented, edge-padded, pos-added Z: (BHW*nseg, seg*D) from x (B,T,H,W,D)
__global__ void build_z(const float* __restrict__ x, const float* __restrict__ pos,
                        float* __restrict__ Z, int seg, int nseg) {
  const int indim = seg * DD;
  const size_t M = (size_t)BB * HH * WW * nseg;
  const size_t total = M * indim;
  for (size_t idx = blockIdx.x * (size_t)blockDim.x + threadIdx.x; idx < total;
       idx += (size_t)gridDim.x * blockDim.x) {
    const size_t row = idx / indim;
    const int k = (int)(idx % indim);
    const int j = (int)(row % nseg);
    const size_t bhw = row / nseg;
    const int p = k / DD;
    const int d = k % DD;
    int t = j * seg + p;
    if (t > TT - 1) t = TT - 1;                 // edge (replicate) pad
    const int b  = (int)(bhw / (HH * WW));
    const int hw = (int)(bhw % (HH * WW));
    Z[idx] = x[(((size_t)b * TT + t) * (HH * WW) + hw) * DD + d]
           + pos[(size_t)j * indim + k];
  }
}

// Tiny per-(bhw, head, j) attention over nseg<=8 segments, hd<=64.
__global__ void attn_kernel(const float* __restrict__ Q, const float* __restrict__ K,
                            const float* __restrict__ V, float* __restrict__ O,
                            int nseg, int hid) {
  const int hd = hid / 8;
  const size_t total = (size_t)BB * HH * WW * 8 * nseg;
  const float scale = rsqrtf((float)hd);
  for (size_t tid = blockIdx.x * (size_t)blockDim.x + threadIdx.x; tid < total;
       tid += (size_t)gridDim.x * blockDim.x) {
    const int j = (int)(tid % nseg);
    const size_t r = tid / nseg;
    const int head = (int)(r % 8);
    const size_t bhw = r / 8;
    const size_t base = bhw * (size_t)nseg * hid + (size_t)head * hd;
    const float* __restrict__ q = Q + base + (size_t)j * hid;
    float sc[8];
    float mx = -3.0e38f;
    for (int m = 0; m < nseg; ++m) {
      const float* __restrict__ kk = K + base + (size_t)m * hid;
      float s = 0.0f;
      for (int d = 0; d < hd; ++d) s += q[d] * kk[d];
      s *= scale;
      sc[m] = s;
      if (s > mx) mx = s;
    }
    float se = 0.0f;
    for (int m = 0; m < nseg; ++m) { sc[m] = expf(sc[m] - mx); se += sc[m]; }
    const float inv = 1.0f / se;
    float* __restrict__ o = O + base + (size_t)j * hid;
    for (int d = 0; d < hd; ++d) {
      float a = 0.0f;
      for (int m = 0; m < nseg; ++m) a += sc[m] * V[base + (size_t)m * hid + d];
      o[d] = a * inv;
    }
  }
}

// acc[b,t,h,w,d] += gates[b,e] * exp(O[bhw*nseg + t/seg][(t%seg)*D + d])
__global__ void combine_expert(const float* __restrict__ O, const float* __restrict__ gates,
                               float* __restrict__ acc, int seg, int nseg, int e) {
  const int indim = seg * DD;
  const size_t total = (size_t)BB * TT * HH * WW * DD;
  for (size_t idx = blockIdx.x * (size_t)blockDim.x + threadIdx.x; idx < total;
       idx += (size_t)gridDim.x * blockDim.x) {
    const int d = (int)(idx % DD);
    const size_t r = idx / DD;
    const size_t hw = r % (HH * WW);
    const size_t r2 = r / (HH * WW);
    const int t = (int)(r2 % TT);
    const int b = (int)(r2 / TT);
    const float g = gates[b * NEXP + e];
    const size_t bhw = (size_t)b * HH * WW + hw;
    const int j = t / seg;
    const int p = t % seg;
    acc[idx] += g * expf(O[(bhw * (size_t)nseg + j) * indim + (size_t)p * DD + d]);
  }
}

__global__ void zero_f32(float* __restrict__ p, size_t n) {
  for (size_t i = blockIdx.x * (size_t)blockDim.x + threadIdx.x; i < n;
       i += (size_t)gridDim.x * blockDim.x) p[i] = 0.0f;
}

__global__ void log_eps(float* __restrict__ p, size_t n) {
  for (size_t i = blockIdx.x * (size_t)blockDim.x + threadIdx.x; i < n;
       i += (size_t)gridDim.x * blockDim.x) {
    float v = p[i];
    if (v == 0.0f) v = COMBINE_EPS;
    p[i] = logf(v);
  }
}

// ---------------------------------------------------------------------------
extern "C" void kernel_launch(void* const* d_in, const int* in_sizes, int n_in,
                              void* d_out, int out_size, void* d_ws, size_t ws_size,
                              hipStream_t stream) {
  (void)in_sizes; (void)n_in; (void)out_size; (void)ws_size;

  const float* x = (const float*)d_in[0];
  const float* conv_w[5]; const float* conv_b[5];
  const float* lng[5];    const float* lnb[5];
  for (int i = 0; i < 5; ++i) {
    conv_w[i] = (const float*)d_in[1 + i];
    conv_b[i] = (const float*)d_in[6 + i];
    lng[i]    = (const float*)d_in[11 + i];
    lnb[i]    = (const float*)d_in[16 + i];
  }
  const float* fuse_w = (const float*)d_in[21];
  const float* fuse_b = (const float*)d_in[22];
  const float* w_gate = (const float*)d_in[23];
  const float* qws[NEXP]; const float* kws[NEXP]; const float* vws[NEXP];
  const float* ows[NEXP]; const float* obs[NEXP]; const float* poss[NEXP];
  for (int e = 0; e < NEXP; ++e) {
    qws[e]  = (const float*)d_in[24 + e];
    kws[e]  = (const float*)d_in[30 + e];
    vws[e]  = (const float*)d_in[36 + e];
    ows[e]  = (const float*)d_in[42 + e];
    obs[e]  = (const float*)d_in[48 + e];
    poss[e] = (const float*)d_in[54 + e];
  }
  float* out = (float*)d_out;

  // Workspace layout (floats)
  const size_t SZ_BIG = (size_t)22 * 1024 * 1024;  // >= 20.97M (seg=5 Z / O)
  const size_t SZ_MED = (size_t)9 * 1024 * 1024;   // >= 8.39M
  float* ws   = (float*)d_ws;
  float* bufZ  = ws;                 // im2col / Z
  float* bufG1 = bufZ  + SZ_BIG;     // act ping / Q
  float* bufG2 = bufG1 + SZ_MED;     // act pong / K / fused h
  float* bufG3 = bufG2 + SZ_MED;     // conv Bt  / V
  float* bufG4 = bufG3 + SZ_MED;     // raw conv GEMM out / attention out
  float* bufO  = bufG4 + SZ_MED;     // expert O projection out
  float* amp   = bufO  + SZ_BIG;     // (B, 8)
  float* gates = amp + 256;          // (B, 6)

  const int GS = 8192;               // grid-stride block count
  const int BT = BB * TT;            // 256

  // ---------------- Gate path: conv stack ----------------
  int Ci = DD, Si = HH;
  const float* act = x;              // x is already NHWC (BT, 32, 32, 64)
  float* pp[2] = {bufG1, bufG2};
  for (int l = 0; l < 5; ++l) {
    const int Co = Ci * 2, So = Si / 2;
    const int M = BT * So * So, N = Co, K = 4 * Ci;
    conv_wtrans<<<1024, 256, 0, stream>>>(conv_w[l], bufG3, Ci, Co);
    im2col2x2<<<GS, 256, 0, stream>>>(act, bufZ, BT, Si, Ci);
    gemm_f32_wmma<<<dim3(N / 64, M / 64), 128, 0, stream>>>(
        bufZ, bufG3, conv_b[l], bufG4, M, N, K);
    ln_gelu<<<M, 256, 0, stream>>>(bufG4, lng[l], lnb[l], pp[l & 1], N);
    act = pp[l & 1];
    Ci = Co; Si = So;
  }
  // act == bufG1 (l=4), shape (256, 2048). Fuse GEMM -> bufG2.
  gemm_f32_wmma<<<dim3(2048 / 64, BT / 64), 128, 0, stream>>>(
      bufG1, fuse_w, fuse_b, bufG2, BT, 2048, 2048);
  fft_amp<<<dim3(BB, 8), 256, 0, stream>>>(bufG2, amp);
  gate_topk<<<1, 32, 0, stream>>>(amp, w_gate, gates);

  // ---------------- Experts ----------------
  const size_t outN = (size_t)BB * TT * HH * WW * DD;
  zero_f32<<<GS, 256, 0, stream>>>(out, outN);

  const int segs[NEXP]  = {16, 8, 5, 4, 3, 2};
  const int nsegs[NEXP] = {1, 2, 4, 4, 6, 8};
  for (int e = 0; e < NEXP; ++e) {
    const int seg = segs[e], n = nsegs[e];
    const int indim = seg * DD;
    const int hid = DD * (seg / 2);
    const int M = BB * HH * WW * n;

    build_z<<<GS, 256, 0, stream>>>(x, poss[e], bufZ, seg, n);
    gemm_f32_wmma<<<dim3(hid / 64, M / 64), 128, 0, stream>>>(
        bufZ, qws[e], nullptr, bufG1, M, hid, indim);
    gemm_f32_wmma<<<dim3(hid / 64, M / 64), 128, 0, stream>>>(
        bufZ, kws[e], nullptr, bufG2, M, hid, indim);
    gemm_f32_wmma<<<dim3(hid / 64, M / 64), 128, 0, stream>>>(
        bufZ, vws[e], nullptr, bufG3, M, hid, indim);
    attn_kernel<<<GS, 256, 0, stream>>>(bufG1, bufG2, bufG3, bufG4, n, hid);
    gemm_f32_wmma<<<dim3(indim / 64, M / 64), 128, 0, stream>>>(
        bufG4, ows[e], obs[e], bufO, M, indim, hid);
    combine_expert<<<GS, 256, 0, stream>>>(bufO, gates, out, seg, n, e);
  }

  log_eps<<<GS, 256, 0, stream>>>(out, outN);
}